// BiMamba2Dv2_31971736552183
// MI455X (gfx1250) — compile-verified
//
#include <hip/hip_runtime.h>
#include <math.h>

// Problem constants (match reference)
#define BB 4
#define HH 48
#define WW 48
#define CC 192           // D_MODEL
#define LL (HH*WW)       // 2304 sequence length
#define BLR (BB*LL)      // 9216 total rows
#define DI 384           // D_INNER
#define NXP 44           // DT_RANK + 2*D_STATE
#define RK 12            // DT_RANK
#define NS 16            // D_STATE

typedef __attribute__((ext_vector_type(2))) float v2f;
typedef __attribute__((ext_vector_type(8))) float v8f;

// ---------------------------------------------------------------------------
// WMMA fp32 GEMM:  C[M=BLR, N] = (accum? C : 0) + A[M,K] * W[N,K]^T
// One wave computes one 16x16 tile, K-stepped by 4 via V_WMMA_F32_16X16X4_F32.
// A-fragment layout (ISA 7.12.2): lanes 0-15 hold K={k0,k0+1}, lanes 16-31
// hold K={k0+2,k0+3}, row M = lane%16. B-fragment mirrored on N.
// NGUARD=false: N is a multiple of 16 -> fully branch-free inner loop
//               (2x global_load_b64 + v_wmma per K-step).
// NGUARD=true : tail lanes clamp their W-row address (always-valid load)
//               and zero the fragment via cndmask -- no exec divergence.
// `reverse` folds per-batch sequence reversal into the A-row addressing.
// ---------------------------------------------------------------------------
template <bool NGUARD>
__global__ __launch_bounds__(128) void wmma_gemm(
    const float* __restrict__ A, int lda,
    const float* __restrict__ Wt, int ldw,
    float* __restrict__ C, int ldc,
    int N, int K, int reverse, int accum)
{
  const int lane = threadIdx.x & 31;
  const int half = lane >> 4;        // 0 or 1
  const int l16  = lane & 15;
  const int mt = (blockIdx.y * blockDim.y + threadIdx.y) * 16;
  const int nt = blockIdx.x * 16;

  int n = nt + l16;
  bool nok = true;
  if (NGUARD) {
    nok = (n < N);
    if (!nok) n = N - 1;             // clamp: loads stay valid, value zeroed below
  }

  int arow = mt + l16;
  if (reverse) {
    int b = arow / LL;
    int l = arow - b * LL;
    arow = b * LL + (LL - 1 - l);
  }
  const float* arp = A  + (size_t)arow * lda + half * 2;
  const float* wrp = Wt + (size_t)n    * ldw + half * 2;

  v8f acc;
#pragma unroll
  for (int r = 0; r < 8; ++r) acc[r] = 0.0f;
  if (accum) {
#pragma unroll
    for (int r = 0; r < 8; ++r) {
      int row = mt + r + half * 8;
      float cv = C[(size_t)row * ldc + n];   // clamped n: valid address
      acc[r] = (!NGUARD || nok) ? cv : 0.0f;
    }
  }

#pragma unroll 4
  for (int k0 = 0; k0 < K; k0 += 4) {
    v2f a = *(const v2f*)(arp + k0);
    v2f b = *(const v2f*)(wrp + k0);
    if (NGUARD && !nok) { b.x = 0.0f; b.y = 0.0f; }   // v_cndmask, no branch
    acc = __builtin_amdgcn_wmma_f32_16x16x4_f32(
        false, a, false, b, (short)0, acc, false, false);
  }

#pragma unroll
  for (int r = 0; r < 8; ++r) {
    int row = mt + r + half * 8;
    if (!NGUARD || nok) C[(size_t)row * ldc + n] = acc[r];
  }
}

// ---------------------------------------------------------------------------
// Reorder x (B,H,W,C) -> useq (B, W*H, C): column-major scan order for pass 0.
// ---------------------------------------------------------------------------
__global__ __launch_bounds__(256) void reorder_wmajor(
    const float* __restrict__ x, float* __restrict__ useq)
{
  int i = blockIdx.x * blockDim.x + threadIdx.x;
  if (i >= BLR * CC) return;
  int c = i % CC;
  int s = (i / CC) % LL;
  int b = i / (CC * LL);
  int w = s / HH;
  int h = s - w * HH;
  useq[i] = x[(((size_t)b * HH + h) * WW + w) * CC + c];
}

// ---------------------------------------------------------------------------
// Depthwise causal conv K=3 + bias + SiLU on the xh half of xz.
// Clamp-then-select (no divergent loads) for the left boundary.
// ---------------------------------------------------------------------------
__global__ __launch_bounds__(256) void conv_silu(
    const float* __restrict__ xz,   // (BLR, 768) for this direction
    const float* __restrict__ cw,   // conv_w base (4,384,3)
    const float* __restrict__ cb,   // conv_b base (4,384)
    float* __restrict__ xc,         // (BLR, 384) for this direction
    int pset)
{
  int idx = blockIdx.x * blockDim.x + threadIdx.x;
  if (idx >= BLR * DI) return;
  int c   = idx % DI;
  int row = idx / DI;
  int b = row / LL;
  int l = row - b * LL;

  float a = cb[pset * DI + c];
  const float* w3 = cw + ((size_t)pset * DI + c) * 3;
#pragma unroll
  for (int k = 0; k < 3; ++k) {
    int ls = l - 2 + k;
    int lc = ls < 0 ? 0 : ls;                              // clamped, valid
    float xv = xz[((size_t)b * LL + lc) * 768 + c];
    a += (ls >= 0 ? xv : 0.0f) * w3[k];                    // cndmask
  }
  float s = a / (1.0f + __expf(-a));   // SiLU
  xc[(size_t)row * DI + c] = s;
}

// ---------------------------------------------------------------------------
// Fused selective scan + gating for BOTH directions (grid.z = dir).
// One thread owns one (b, channel): h[16] and A[16] live in VGPRs.
// delta_raw was written into the xh half of xz by the dt_proj GEMM;
// z lives in the upper half of xz. Gated output overwrites xc in place.
// ---------------------------------------------------------------------------
__global__ __launch_bounds__(128) void scan_gate(
    const float* __restrict__ xz,    // (2, BLR, 768)
    const float* __restrict__ xdbl,  // (2, BLR, 44)
    float*       __restrict__ xc,    // (2, BLR, 384) in: xc, out: y*silu(z)
    const float* __restrict__ dtb,   // (4, 384)
    const float* __restrict__ Alog,  // (4, 384, 16)
    const float* __restrict__ Dp,    // (4, 384)
    int pass)
{
  int dir = blockIdx.z;
  int t = blockIdx.x * blockDim.x + threadIdx.x;
  if (t >= BB * DI) return;
  int c = t % DI;
  int b = t / DI;
  int pset = pass * 2 + dir;

  float A[NS], h[NS];
#pragma unroll
  for (int nn = 0; nn < NS; ++nn) {
    A[nn] = -__expf(Alog[((size_t)pset * DI + c) * NS + nn]);
    h[nn] = 0.0f;
  }
  const float bias = dtb[pset * DI + c];
  const float dpar = Dp[pset * DI + c];

  const float* xzp = xz   + (size_t)dir * BLR * 768 + (size_t)b * LL * 768;
  const float* xdp = xdbl + (size_t)dir * BLR * NXP + (size_t)b * LL * NXP;
  float*       xcp = xc   + (size_t)dir * BLR * DI  + (size_t)b * LL * DI;

  for (int l = 0; l < LL; ++l) {
    float draw  = xzp[(size_t)l * 768 + c] + bias;
    float delta = (draw > 20.0f) ? draw : log1pf(__expf(draw));   // softplus
    float u  = xcp[(size_t)l * DI + c];
    float du = delta * u;

    const float4* bm = (const float4*)(xdp + (size_t)l * NXP + RK);        // B: cols 12..27
    const float4* cm = (const float4*)(xdp + (size_t)l * NXP + RK + NS);   // C: cols 28..43

    float y = 0.0f;
#pragma unroll
    for (int q = 0; q < 4; ++q) {
      float4 bq = bm[q];
      float4 cq = cm[q];
      float e0 = __expf(delta * A[4*q + 0]);
      float e1 = __expf(delta * A[4*q + 1]);
      float e2 = __expf(delta * A[4*q + 2]);
      float e3 = __expf(delta * A[4*q + 3]);
      h[4*q + 0] = e0 * h[4*q + 0] + du * bq.x;  y += h[4*q + 0] * cq.x;
      h[4*q + 1] = e1 * h[4*q + 1] + du * bq.y;  y += h[4*q + 1] * cq.y;
      h[4*q + 2] = e2 * h[4*q + 2] + du * bq.z;  y += h[4*q + 2] * cq.z;
      h[4*q + 3] = e3 * h[4*q + 3] + du * bq.w;  y += h[4*q + 3] * cq.w;
    }
    y += u * dpar;

    float zz = xzp[(size_t)l * 768 + DI + c];
    float g  = zz / (1.0f + __expf(-zz));        // SiLU(z)
    xcp[(size_t)l * DI + c] = y * g;
  }
}

// ---------------------------------------------------------------------------
// LayerNorm(remapped pass-0 output) + residual(x) -> useq for pass 1.
// One wave per 192-channel row; wave32 shuffle reduction.
// ---------------------------------------------------------------------------
__global__ __launch_bounds__(256) void ln_residual(
    const float* __restrict__ yw,    // (B, LL w-major, 192)
    const float* __restrict__ x,     // (B,H,W,192)
    const float* __restrict__ nw, const float* __restrict__ nb,
    float* __restrict__ useq)        // (B, H*W, 192)
{
  int lane = threadIdx.x;                       // 0..31
  int row = blockIdx.x * blockDim.y + threadIdx.y;
  if (row >= BLR) return;
  int b = row / LL;
  int r = row - b * LL;
  int h = r / WW;
  int w = r - h * WW;
  const float* yrow = yw + ((size_t)b * LL + (size_t)w * HH + h) * CC;

  float v[6];
  float s = 0.0f, s2 = 0.0f;
#pragma unroll
  for (int j = 0; j < 6; ++j) {
    v[j] = yrow[lane + 32 * j];
    s  += v[j];
    s2 += v[j] * v[j];
  }
#pragma unroll
  for (int m = 16; m >= 1; m >>= 1) {
    s  += __shfl_xor(s,  m, 32);
    s2 += __shfl_xor(s2, m, 32);
  }
  float mean = s * (1.0f / CC);
  float var  = s2 * (1.0f / CC) - mean * mean;
  float rstd = rsqrtf(var + 1e-5f);

  const float* xrow = x    + (size_t)row * CC;
  float*       orow = useq + (size_t)row * CC;
#pragma unroll
  for (int j = 0; j < 6; ++j) {
    int cch = lane + 32 * j;
    orow[cch] = (v[j] - mean) * rstd * nw[cch] + nb[cch] + xrow[cch];
  }
}

// ---------------------------------------------------------------------------
extern "C" void kernel_launch(void* const* d_in, const int* in_sizes, int n_in,
                              void* d_out, int out_size, void* d_ws, size_t ws_size,
                              hipStream_t stream) {
  (void)in_sizes; (void)n_in; (void)out_size; (void)ws_size;
  const float* x     = (const float*)d_in[0];
  const float* inW   = (const float*)d_in[1];
  const float* convW = (const float*)d_in[2];
  const float* convB = (const float*)d_in[3];
  const float* xpW   = (const float*)d_in[4];
  const float* dtW   = (const float*)d_in[5];
  const float* dtB   = (const float*)d_in[6];
  const float* Alog  = (const float*)d_in[7];
  const float* Dp    = (const float*)d_in[8];
  const float* outW  = (const float*)d_in[9];
  const float* nw    = (const float*)d_in[10];
  const float* nb    = (const float*)d_in[11];
  float* out = (float*)d_out;

  float* ws   = (float*)d_ws;
  float* useq = ws;                                 // BLR*CC
  float* yw   = useq + (size_t)BLR * CC;            // BLR*CC
  float* xz   = yw   + (size_t)BLR * CC;            // 2*BLR*768
  float* xc   = xz   + (size_t)2 * BLR * 768;       // 2*BLR*384
  float* xdbl = xc   + (size_t)2 * BLR * DI;        // 2*BLR*44

  // pass-0 scan order (B, W*H, C)
  reorder_wmajor<<<dim3((BLR * CC + 255) / 256), dim3(256), 0, stream>>>(x, useq);

  const dim3 gblk(32, 4);                 // 4 wave32 waves; one 16x16 tile each
  const int MT = BLR / 64;                // 144 M-tile groups

  for (int pass = 0; pass < 2; ++pass) {
    float* dst = (pass == 0) ? yw : out;

    for (int dir = 0; dir < 2; ++dir) {
      int pset = pass * 2 + dir;
      float* xzd = xz   + (size_t)dir * BLR * 768;
      float* xcd = xc   + (size_t)dir * BLR * DI;
      float* xdd = xdbl + (size_t)dir * BLR * NXP;

      // in_proj: (BLR,192)x(192,768); dir 1 reads the sequence reversed
      wmma_gemm<false><<<dim3(768 / 16, MT), gblk, 0, stream>>>(
          useq, CC, inW + (size_t)pset * 768 * CC, CC,
          xzd, 768, 768, CC, dir, 0);

      // causal conv + SiLU on xh half
      conv_silu<<<dim3((BLR * DI + 255) / 256), dim3(256), 0, stream>>>(
          xzd, convW, convB, xcd, pset);

      // x_proj: (BLR,384)x(384,44) -- only GEMM with an N tail
      wmma_gemm<true><<<dim3((NXP + 15) / 16, MT), gblk, 0, stream>>>(
          xcd, DI, xpW + (size_t)pset * NXP * DI, DI,
          xdd, NXP, NXP, DI, 0, 0);

      // dt_proj: (BLR,12)x(12,384) -> delta_raw into the consumed xh half of xz
      wmma_gemm<false><<<dim3(DI / 16, MT), gblk, 0, stream>>>(
          xdd, NXP, dtW + (size_t)pset * DI * RK, RK,
          xzd, 768, DI, RK, 0, 0);
    }

    // both directions scanned concurrently (grid.z = dir); gated in place
    scan_gate<<<dim3((BB * DI + 127) / 128, 1, 2), dim3(128), 0, stream>>>(
        xz, xdbl, xc, dtB, Alog, Dp, pass);

    // out_proj: dir 0 writes, dir 1 accumulates (bimamba sum)
    for (int dir = 0; dir < 2; ++dir) {
      int pset = pass * 2 + dir;
      wmma_gemm<false><<<dim3(CC / 16, MT), gblk, 0, stream>>>(
          xc + (size_t)dir * BLR * DI, DI,
          outW + (size_t)pset * CC * DI, DI,
          dst, CC, CC, DI, 0, dir);
    }

    if (pass == 0) {
      // LN(remap to row-major) + residual -> pass-1 input sequence
      ln_residual<<<dim3(BLR / 8), dim3(32, 8), 0, stream>>>(
          yw, x, nw, nb, useq);
    }
  }
}